// GCNLayer_33809982554664
// MI455X (gfx1250) — compile-verified
//
#include <hip/hip_runtime.h>
#include <hip/hip_bf16.h>

typedef __attribute__((ext_vector_type(2))) float v2f;
typedef __attribute__((ext_vector_type(8))) float v8f;

#define IN_FEATS   128
#define OUT_FEATS  128
#define LDS_STRIDE 132   // pad 128 -> 132 floats: rows land on distinct LDS banks

// ---------------------------------------------------------------------------
// Kernel 1: zero the accumulator (d_out doubles as h then becomes the output)
// ---------------------------------------------------------------------------
__global__ __launch_bounds__(256) void gcn_zero_kernel(float4* __restrict__ p, int n4) {
    int i = blockIdx.x * 256 + threadIdx.x;
    if (i < n4) p[i] = make_float4(0.f, 0.f, 0.f, 0.f);
}

// ---------------------------------------------------------------------------
// Kernel 2: edge scatter-add.  One wave32 per edge; each lane moves 4 floats
// (float4 load -> 4x global_atomic_add_f32, relaxed/agent scope so it lowers
// to the native no-return FP atomic, not a CAS loop).  feat/h are L2-resident
// (51.2 MB each vs 192 MB L2), so this is atomic-throughput bound, not HBM.
// ---------------------------------------------------------------------------
__global__ __launch_bounds__(256) void gcn_scatter_kernel(
    const float* __restrict__ feat,
    const int*   __restrict__ edge_src,
    const int*   __restrict__ edge_dst,
    float*       __restrict__ h,
    int n_edges)
{
    int edge = blockIdx.x * 8 + (threadIdx.x >> 5);   // 8 edges per 256-thread block
    if (edge >= n_edges) return;
    int lane = threadIdx.x & 31;

    int s = edge_src[edge];
    int d = edge_dst[edge];

    const float4 v = *(const float4*)(feat + (size_t)s * IN_FEATS + lane * 4);
    float* dst = h + (size_t)d * IN_FEATS + lane * 4;
    __hip_atomic_fetch_add(dst + 0, v.x, __ATOMIC_RELAXED, __HIP_MEMORY_SCOPE_AGENT);
    __hip_atomic_fetch_add(dst + 1, v.y, __ATOMIC_RELAXED, __HIP_MEMORY_SCOPE_AGENT);
    __hip_atomic_fetch_add(dst + 2, v.z, __ATOMIC_RELAXED, __HIP_MEMORY_SCOPE_AGENT);
    __hip_atomic_fetch_add(dst + 3, v.w, __ATOMIC_RELAXED, __HIP_MEMORY_SCOPE_AGENT);
}

// ---------------------------------------------------------------------------
// Kernel 3: in-place linear layer  out[m,:] = h[m,:] @ W^T + bias
// Block = 256 threads = 8 waves; block owns 16 rows (exclusive), wave w owns
// the 16-wide output-column tile [16w, 16w+16).  h tile is staged through LDS
// so the in-place overwrite is race-free.  K=128 swept in steps of 4 via
// V_WMMA_F32_16X16X4_F32 (f32 in / f32 accumulate, matches reference math).
// ---------------------------------------------------------------------------
__global__ __launch_bounds__(256) void gcn_gemm_wmma_kernel(
    float*       __restrict__ hout,   // in: h accumulator, out: final result (in place)
    const float* __restrict__ W,      // [OUT_FEATS, IN_FEATS] row-major
    const float* __restrict__ bias,   // [OUT_FEATS]
    int n_nodes)
{
    __shared__ float tile[16 * LDS_STRIDE];

    const int tid     = threadIdx.x;
    const int rowBase = blockIdx.x * 16;
    const bool fullTile = (rowBase + 16 <= n_nodes);  // uniform across the block

    // Cooperative load of the block's 16x128 h tile into LDS.
    for (int i = tid; i < 16 * IN_FEATS; i += 256) {
        int r = i >> 7;          // /128
        int c = i & (IN_FEATS - 1);
        int row = rowBase + r;
        if (row >= n_nodes) row = n_nodes - 1;   // clamp (n_nodes%16==0 in practice)
        tile[r * LDS_STRIDE + c] = hout[(size_t)row * IN_FEATS + c];
    }
    __syncthreads();

    const int wave  = tid >> 5;        // 0..7 -> output-column tile
    const int lane  = tid & 31;
    const int lhalf = lane >> 4;       // 0 or 1
    const int l16   = lane & 15;
    const int nCol  = wave * 16 + l16; // output feature this lane owns in B/C/D

    // C/D layout: lane l, VGPR v  ->  (m = v + 8*lhalf, n = l16).
    // Initialize C with bias[n] so the bias add is fused into the WMMA chain.
    v8f acc;
    {
        float bv = bias[nCol];
#pragma unroll
        for (int v = 0; v < 8; ++v) acc[v] = bv;
    }

    // A layout (16x4 f32): lane l, reg j -> A[l16, k = kc + 2*lhalf + j]
    // B layout (4x16 f32): lane l, reg j -> B[k = kc + 2*lhalf + j, l16],
    //   with B[k,n] = W[n,k]  (both are contiguous float pairs along k).
    const float* wrow = W + (size_t)nCol * IN_FEATS;
#pragma unroll 4
    for (int kc = 0; kc < IN_FEATS; kc += 4) {
        const int kk = kc + 2 * lhalf;

        v2f a;
        a[0] = tile[l16 * LDS_STRIDE + kk];
        a[1] = tile[l16 * LDS_STRIDE + kk + 1];

        v2f b;
        b[0] = wrow[kk];
        b[1] = wrow[kk + 1];

        // (neg_a, A, neg_b, B, c_mod, C, reuse_a, reuse_b)
        acc = __builtin_amdgcn_wmma_f32_16x16x4_f32(
            false, a, false, b, (short)0, acc, false, false);
    }

    // Write the 16x16 D tile back in place (rows are block-exclusive).
    float* outBase = hout + (size_t)(rowBase + 8 * lhalf) * OUT_FEATS + nCol;
    if (fullTile) {
        // Branch-free fast path (always taken when n_nodes % 16 == 0).
#pragma unroll
        for (int v = 0; v < 8; ++v)
            outBase[(size_t)v * OUT_FEATS] = acc[v];
    } else {
#pragma unroll
        for (int v = 0; v < 8; ++v) {
            int m = rowBase + v + 8 * lhalf;
            if (m < n_nodes)
                hout[(size_t)m * OUT_FEATS + nCol] = acc[v];
        }
    }
}

// ---------------------------------------------------------------------------
// Launch: zero -> scatter -> in-place WMMA GEMM (stream-ordered, no d_ws use)
// ---------------------------------------------------------------------------
extern "C" void kernel_launch(void* const* d_in, const int* in_sizes, int n_in,
                              void* d_out, int out_size, void* d_ws, size_t ws_size,
                              hipStream_t stream) {
    const float* feat     = (const float*)d_in[0];
    const int*   edge_src = (const int*)d_in[1];
    const int*   edge_dst = (const int*)d_in[2];
    const float* W        = (const float*)d_in[3];
    const float* bias     = (const float*)d_in[4];
    float*       out      = (float*)d_out;

    const int n_nodes = in_sizes[0] / IN_FEATS;   // 100000
    const int n_edges = in_sizes[1];              // 640000

    // 1) zero the accumulator region of d_out
    const int n4 = (n_nodes * IN_FEATS) / 4;
    gcn_zero_kernel<<<(n4 + 255) / 256, 256, 0, stream>>>((float4*)out, n4);

    // 2) scatter-add messages: one wave per edge, 8 edges per block
    gcn_scatter_kernel<<<(n_edges + 7) / 8, 256, 0, stream>>>(
        feat, edge_src, edge_dst, out, n_edges);

    // 3) in-place linear layer with f32 WMMA; 16 rows per block
    gcn_gemm_wmma_kernel<<<(n_nodes + 15) / 16, 256, 0, stream>>>(
        out, W, bias, n_nodes);
}